// SybilGATNetwork_76046690943379
// MI455X (gfx1250) — compile-verified
//
#include <hip/hip_runtime.h>
#include <hip/hip_bf16.h>
#include <math.h>

// ---------------------------------------------------------------------------
// SybilGATNetwork for MI455X (gfx1250, wave32).
// GEMMs (8x [50000x128]x[128x128]) via V_WMMA_F32_16X16X4_F32 (fp32 matches
// reference precision; GEMM flops ~13 GF are not the bottleneck).  Edge
// softmax/scatter phase is the roofline limiter (~2.7 GB, L2-resident since
// all [N,128] buffers are 25.6 MB << 192 MB L2).
// Weights staged in LDS as K-pair-major float2 so every WMMA B-fragment is a
// single contiguous ds_load_b64 (no VGPR re-pairing moves in the hot loop).
// ---------------------------------------------------------------------------

#define HIDDIM 128

typedef __attribute__((ext_vector_type(2))) float v2f;
typedef __attribute__((ext_vector_type(8))) float v8f;

// Order-preserving float <-> uint mapping for atomic max on floats.
__device__ __forceinline__ unsigned encOrd(float f) {
  unsigned b = __float_as_uint(f);
  return (b & 0x80000000u) ? ~b : (b | 0x80000000u);
}
__device__ __forceinline__ float decOrd(unsigned k) {
  unsigned b = (k & 0x80000000u) ? (k & 0x7fffffffu) : ~k;
  return __uint_as_float(b);
}
#define ENC_NEG_INF 0x007FFFFFu  // encOrd(-inf)

__global__ void fill_u32(unsigned* __restrict__ p, unsigned v, long n) {
  long i = (long)blockIdx.x * blockDim.x + threadIdx.x;
  if (i < n) p[i] = v;
}

// ---------------------------------------------------------------------------
// C[row, 0..127] = act(A[row,0..127] @ W^T + bias) + res   (W is [128,128])
// One wave -> 16 rows x 128 cols.  K loop in steps of 4 via WMMA f32 16x16x4.
// LDS layout: sP[p*128+n] = float2{ W[n][2p], W[n][2p+1] }, p = K-pair index.
// A lane needing K-pair (k/2 + half) for column n does ONE ds_load_b64 that
// lands directly in the even-aligned VGPR pair consumed by the WMMA.
// ---------------------------------------------------------------------------
__global__ __launch_bounds__(128) void gemm_n128_wmma(
    const float* __restrict__ A, const float* __restrict__ W,
    const float* __restrict__ bias, const float* __restrict__ res,
    float* __restrict__ C, int nrows, int relu_act)
{
  __shared__ v2f sP[(HIDDIM / 2) * HIDDIM];  // 64 K-pairs x 128 cols = 64 KB
  for (int i = threadIdx.x; i < (HIDDIM / 2) * HIDDIM; i += blockDim.x) {
    int p = i >> 7, n = i & 127;
    sP[i] = *(const v2f*)(W + n * HIDDIM + 2 * p);  // {W[n][2p], W[n][2p+1]}
  }
  __syncthreads();

  const int wave = threadIdx.x >> 5;
  const int lane = threadIdx.x & 31;
  const int half = lane >> 4;   // 0: K pair {2p}   1: K pair {2p+1}
  const int l16  = lane & 15;
  const int rowBase = (blockIdx.x * 4 + wave) * 16;

  int aRow = rowBase + l16;
  if (aRow >= nrows) aRow = nrows - 1;            // clamp loads; stores guarded
  const float* aPtr = A + (long)aRow * HIDDIM + half * 2;

  v8f acc[8];
#pragma unroll
  for (int t = 0; t < 8; ++t)
#pragma unroll
    for (int r = 0; r < 8; ++r) acc[t][r] = 0.0f;

  for (int k = 0; k < HIDDIM; k += 4) {
    v2f a = *(const v2f*)(aPtr + k);              // A[m, k+2*half .. +1]
    const v2f* wp = sP + (k / 2 + half) * HIDDIM + l16;
#pragma unroll
    for (int t = 0; t < 8; ++t) {                 // 8 N-tiles of 16
      v2f b = wp[t * 16];                         // {B[k+2h][n], B[k+2h+1][n]}
      acc[t] = __builtin_amdgcn_wmma_f32_16x16x4_f32(
          false, a, false, b, (short)0, acc[t], false, false);
    }
  }

  // C/D layout: VGPR r -> M = r + 8*half, N = l16 (+16*t)
#pragma unroll
  for (int t = 0; t < 8; ++t) {
    int col = t * 16 + l16;
    float bv = bias ? bias[col] : 0.0f;
#pragma unroll
    for (int r = 0; r < 8; ++r) {
      int row = rowBase + half * 8 + r;
      if (row < nrows) {
        float v = acc[t][r] + bv;
        if (relu_act) v = fmaxf(v, 0.0f);
        if (res) v += res[(long)row * HIDDIM + col];
        C[(long)row * HIDDIM + col] = v;
      }
    }
  }
}

// es[n,h] = sum_d xp[n,h,d]*As[h,d];  ed likewise.
__global__ void attn_scores(const float* __restrict__ xp,
                            const float* __restrict__ As, const float* __restrict__ Ad,
                            float* __restrict__ es, float* __restrict__ ed,
                            int n, int heads) {
  long idx = (long)blockIdx.x * blockDim.x + threadIdx.x;
  if (idx >= (long)n * heads) return;
  int node = (int)(idx / heads), h = (int)(idx - (long)node * heads);
  int d = HIDDIM / heads;
  const float* xr = xp + (long)node * HIDDIM + h * d;
  const float* as = As + h * d;
  const float* ad = Ad + h * d;
  float s = 0.f, t = 0.f;
  for (int j = 0; j < d; ++j) { float v = xr[j]; s = fmaf(v, as[j], s); t = fmaf(v, ad[j], t); }
  es[idx] = s; ed[idx] = t;
}

// Pass 1: e = leaky_relu(es[src]+ed[dst]); segment max into mEnc[dst].
__global__ void edge_logits(const int* __restrict__ src, const int* __restrict__ dst,
                            const float* __restrict__ es, const float* __restrict__ ed,
                            float* __restrict__ eBuf, unsigned* __restrict__ mEnc,
                            int E, int n, int heads) {
  long idx = (long)blockIdx.x * blockDim.x + threadIdx.x;
  if (idx >= (long)(E + n) * heads) return;
  int e = (int)(idx / heads), h = (int)(idx - (long)e * heads);
  int s = (e < E) ? src[e] : (e - E);     // self loops appended
  int d = (e < E) ? dst[e] : (e - E);
  float v = es[(long)s * heads + h] + ed[(long)d * heads + h];
  v = (v > 0.f) ? v : 0.2f * v;
  eBuf[idx] = v;
  atomicMax(&mEnc[(long)d * heads + h], encOrd(v));
}

// Pass 2: ex = exp(e - m[dst]); segment sum into denom[dst].
__global__ void edge_exp(const int* __restrict__ src, const int* __restrict__ dst,
                         float* __restrict__ eBuf, const unsigned* __restrict__ mEnc,
                         float* __restrict__ denom, int E, int n, int heads) {
  long idx = (long)blockIdx.x * blockDim.x + threadIdx.x;
  if (idx >= (long)(E + n) * heads) return;
  int e = (int)(idx / heads), h = (int)(idx - (long)e * heads);
  int d = (e < E) ? dst[e] : (e - E);
  float m = decOrd(mEnc[(long)d * heads + h]);
  float ex = expf(eBuf[idx] - m);
  eBuf[idx] = ex;
  atomicAdd(&denom[(long)d * heads + h], ex);
}

// Pass 3: aggr[dst] += xp[src] * alpha.  One wave per edge, 4 floats/lane.
__global__ void edge_scatter(const int* __restrict__ src, const int* __restrict__ dst,
                             const float* __restrict__ xp, const float* __restrict__ eBuf,
                             const float* __restrict__ denom, float* __restrict__ aggr,
                             int E, int n, int heads) {
  long wid = ((long)blockIdx.x * blockDim.x + threadIdx.x) >> 5;
  int lane = threadIdx.x & 31;
  if (wid >= (long)(E + n)) return;
  int e = (int)wid;
  int s = (e < E) ? src[e] : (e - E);
  int d = (e < E) ? dst[e] : (e - E);
  int dph = HIDDIM / heads;
  const float* xs = xp + (long)s * HIDDIM;
  float* od = aggr + (long)d * HIDDIM;
#pragma unroll
  for (int j = 0; j < 4; ++j) {
    int f = lane * 4 + j;
    int h = f / dph;
    float alpha = eBuf[(long)e * heads + h] / denom[(long)d * heads + h];
    atomicAdd(&od[f], xs[f] * alpha);
  }
}

// h = elu(aggr + bg) (+ residual)
__global__ void gat_finalize(const float* __restrict__ aggr, const float* __restrict__ bg,
                             const float* __restrict__ res, float* __restrict__ hOut, int n) {
  long idx = (long)blockIdx.x * blockDim.x + threadIdx.x;
  if (idx >= (long)n * HIDDIM) return;
  int f = (int)(idx & (HIDDIM - 1));
  float v = aggr[idx] + bg[f];
  v = (v > 0.f) ? v : expm1f(v);
  if (res) v += res[idx];
  hOut[idx] = v;
}

// Per-graph sum / sum-of-squares / max / count via atomics.
__global__ void pool_scatter(const float* __restrict__ enh, const int* __restrict__ batch,
                             float* __restrict__ s, float* __restrict__ ss,
                             unsigned* __restrict__ mxEnc, float* __restrict__ cnt, int n) {
  long idx = (long)blockIdx.x * blockDim.x + threadIdx.x;
  if (idx >= (long)n * HIDDIM) return;
  int node = (int)(idx >> 7);
  int f = (int)(idx & 127);
  int g = batch[node];
  float v = enh[idx];
  atomicAdd(&s[(long)g * HIDDIM + f], v);
  atomicAdd(&ss[(long)g * HIDDIM + f], v * v);
  atomicMax(&mxEnc[(long)g * HIDDIM + f], encOrd(v));
  if (f == 0) atomicAdd(&cnt[g], 1.0f);
}

// Build g=[mean|max|std] then 384->128->64->2 MLP.  One block per graph.
__global__ __launch_bounds__(128) void pool_classify(
    const float* __restrict__ s, const float* __restrict__ ss,
    const unsigned* __restrict__ mxEnc, const float* __restrict__ cnt,
    const float* __restrict__ Wc1, const float* __restrict__ bc1,
    const float* __restrict__ Wc2, const float* __restrict__ bc2,
    const float* __restrict__ Wc3, const float* __restrict__ bc3,
    float* __restrict__ out) {
  __shared__ float gv[3 * HIDDIM];
  __shared__ float h1s[HIDDIM];
  __shared__ float h2s[64];
  int g = blockIdx.x;
  int j = threadIdx.x;
  float c  = cnt[g];
  float cc = fmaxf(c, 1.0f);
  float mean = s[(long)g * HIDDIM + j] / cc;
  float mx   = decOrd(mxEnc[(long)g * HIDDIM + j]);
  float var  = (ss[(long)g * HIDDIM + j] - cc * mean * mean) / fmaxf(c - 1.0f, 1.0f);
  float sd   = (c > 1.0f) ? sqrtf(fmaxf(var, 0.0f)) : 0.0f;
  gv[j] = mean; gv[HIDDIM + j] = mx; gv[2 * HIDDIM + j] = sd;
  __syncthreads();
  {
    float acc = bc1[j];
    for (int k = 0; k < 3 * HIDDIM; ++k) acc = fmaf(gv[k], Wc1[j * 3 * HIDDIM + k], acc);
    h1s[j] = fmaxf(acc, 0.0f);
  }
  __syncthreads();
  if (j < 64) {
    float acc = bc2[j];
    for (int k = 0; k < HIDDIM; ++k) acc = fmaf(h1s[k], Wc2[j * HIDDIM + k], acc);
    h2s[j] = fmaxf(acc, 0.0f);
  }
  __syncthreads();
  if (j < 2) {
    float acc = bc3[j];
    for (int k = 0; k < 64; ++k) acc = fmaf(h2s[k], Wc3[j * 64 + k], acc);
    out[g * 2 + j] = acc;
  }
}

extern "C" void kernel_launch(void* const* d_in, const int* in_sizes, int n_in,
                              void* d_out, int out_size, void* d_ws, size_t ws_size,
                              hipStream_t stream) {
  const float* x    = (const float*)d_in[0];
  const int*   eidx = (const int*)d_in[1];
  const int*   batch= (const int*)d_in[2];
  const float* Wi1  = (const float*)d_in[3];
  const float* bi1  = (const float*)d_in[4];
  const float* Wi2  = (const float*)d_in[5];
  const float* bi2  = (const float*)d_in[6];
  const float* Wg[4] = {(const float*)d_in[7],  (const float*)d_in[11],
                        (const float*)d_in[15], (const float*)d_in[19]};
  const float* As[4] = {(const float*)d_in[8],  (const float*)d_in[12],
                        (const float*)d_in[16], (const float*)d_in[20]};
  const float* Ad[4] = {(const float*)d_in[9],  (const float*)d_in[13],
                        (const float*)d_in[17], (const float*)d_in[21]};
  const float* bg[4] = {(const float*)d_in[10], (const float*)d_in[14],
                        (const float*)d_in[18], (const float*)d_in[22]};
  const float* We1 = (const float*)d_in[23];
  const float* be1 = (const float*)d_in[24];
  const float* We2 = (const float*)d_in[25];
  const float* be2 = (const float*)d_in[26];
  const float* Wc1 = (const float*)d_in[27];
  const float* bc1 = (const float*)d_in[28];
  const float* Wc2 = (const float*)d_in[29];
  const float* bc2 = (const float*)d_in[30];
  const float* Wc3 = (const float*)d_in[31];
  const float* bc3 = (const float*)d_in[32];

  const int E = in_sizes[1] / 2;
  const int n = in_sizes[2];
  const int* src  = eidx;       // edge_index[0]
  const int* dstp = eidx + E;   // edge_index[1]

  // workspace layout (floats)
  float* base = (float*)d_ws;
  size_t off = 0;
  auto alloc = [&](size_t nf) { float* r = base + off; off += nf; return r; };
  float*    H0    = alloc((size_t)n * HIDDIM);
  float*    H1    = alloc((size_t)n * HIDDIM);
  float*    XP    = alloc((size_t)n * HIDDIM);
  float*    AGG   = alloc((size_t)n * HIDDIM);
  float*    T1    = alloc((size_t)n * HIDDIM);
  float*    ES    = alloc((size_t)n * 8);
  float*    ED    = alloc((size_t)n * 8);
  unsigned* mEnc  = (unsigned*)alloc((size_t)n * 8);
  float*    denom = alloc((size_t)n * 8);
  float*    eBuf  = alloc((size_t)(E + n) * 8);
  float*    sPool = alloc(64 * HIDDIM);
  float*    ssPool= alloc(64 * HIDDIM);
  unsigned* mxEnc = (unsigned*)alloc(64 * HIDDIM);
  float*    cntP  = alloc(64);
  (void)ws_size; (void)n_in; (void)out_size;

  const int gemmGrid = (n + 63) / 64;
  auto blks = [](long t) { return (unsigned)((t + 255) / 256); };

  // input_proj: h = relu(x@Wi1.T+bi1) @ Wi2.T + bi2
  gemm_n128_wmma<<<gemmGrid, 128, 0, stream>>>(x,  Wi1, bi1, nullptr, T1, n, 1);
  gemm_n128_wmma<<<gemmGrid, 128, 0, stream>>>(T1, Wi2, bi2, nullptr, H0, n, 0);

  float* hc = H0; float* hn = H1;
  const int headsArr[4] = {8, 8, 8, 1};
  for (int l = 0; l < 4; ++l) {
    const int heads = headsArr[l];
    const long nh = (long)n * heads;
    const long eh = (long)(E + n) * heads;
    const long nf = (long)n * HIDDIM;
    gemm_n128_wmma<<<gemmGrid, 128, 0, stream>>>(hc, Wg[l], nullptr, nullptr, XP, n, 0);
    attn_scores<<<blks(nh), 256, 0, stream>>>(XP, As[l], Ad[l], ES, ED, n, heads);
    fill_u32<<<blks(nh), 256, 0, stream>>>(mEnc, ENC_NEG_INF, nh);
    edge_logits<<<blks(eh), 256, 0, stream>>>(src, dstp, ES, ED, eBuf, mEnc, E, n, heads);
    fill_u32<<<blks(nh), 256, 0, stream>>>((unsigned*)denom, 0u, nh);
    edge_exp<<<blks(eh), 256, 0, stream>>>(src, dstp, eBuf, mEnc, denom, E, n, heads);
    fill_u32<<<blks(nf), 256, 0, stream>>>((unsigned*)AGG, 0u, nf);
    edge_scatter<<<blks((long)(E + n) * 32), 256, 0, stream>>>(src, dstp, XP, eBuf, denom, AGG, E, n, heads);
    gat_finalize<<<blks(nf), 256, 0, stream>>>(AGG, bg[l], (l > 0) ? hc : nullptr, hn, n);
    float* t = hc; hc = hn; hn = t;
  }

  // enh = h + relu(h@We1.T+be1)@We2.T+be2
  gemm_n128_wmma<<<gemmGrid, 128, 0, stream>>>(hc, We1, be1, nullptr, T1, n, 1);
  gemm_n128_wmma<<<gemmGrid, 128, 0, stream>>>(T1, We2, be2, hc, hn, n, 0);
  float* ENH = hn;

  // pooling + classifier
  fill_u32<<<blks(64 * HIDDIM), 256, 0, stream>>>((unsigned*)sPool,  0u, 64 * HIDDIM);
  fill_u32<<<blks(64 * HIDDIM), 256, 0, stream>>>((unsigned*)ssPool, 0u, 64 * HIDDIM);
  fill_u32<<<blks(64 * HIDDIM), 256, 0, stream>>>(mxEnc, ENC_NEG_INF, 64 * HIDDIM);
  fill_u32<<<1, 64, 0, stream>>>((unsigned*)cntP, 0u, 64);
  pool_scatter<<<blks((long)n * HIDDIM), 256, 0, stream>>>(ENH, batch, sPool, ssPool, mxEnc, cntP, n);
  pool_classify<<<64, 128, 0, stream>>>(sPool, ssPool, mxEnc, cntP,
                                        Wc1, bc1, Wc2, bc2, Wc3, bc3, (float*)d_out);
}